// _PLACModule_56384330662109
// MI455X (gfx1250) — compile-verified
//
#include <hip/hip_runtime.h>

// PLAC (piecewise-linear activation, Q16.16) for MI455X / gfx1250.
//
// Roofline: 134.2M elems * 8B = 1.07 GB single-touch traffic -> ~46 us at
// 23.3 TB/s, i.e. ~6 elems/WGP/cycle. A WGP does 4x32 VALU lane-ops/cycle,
// so the ALU budget is ~21 VALU/element. A 15-compare linear searchsorted
// (~30 VALU/elem) busts that budget, so the bucketize is a 4-step binary
// search: step 1 compares against SGPR-resident bp[7]; steps 2-4 fetch the
// data-dependent breakpoint with ds_bpermute_b32 (wave32 crossbar, no LDS
// bank traffic), keeping ~23 VALU + 4 DS per element. The per-segment
// coefficients come from one conflict-free ds_load_b64 (16 x 8B LDS table,
// entries {b - sm, e | sm<<31} so sign application is xor+add).
// No matmul structure => no WMMA; global traffic uses 128-bit NT loads and
// stores (both streams are single-touch and larger than the 192MB L2).

typedef float v4f __attribute__((ext_vector_type(4)));

#define NBP 15
#define NSEG 16
#define INV_SCALE 1.52587890625e-05f  // 2^-16, exact

// Fast path: requires full EXEC (uses ds_bpermute reading lanes 0..14).
// bp7: wave-uniform SGPR copy of breakpoints[7].
// bpl: per-lane register file of breakpoints (lane i holds bp[min(i,14)]).
// tab: 16-entry LDS coefficient table.
__device__ __forceinline__ float plac_fast(float xf, int bp7, int bpl,
                                           const int2* __restrict__ tab) {
    // v_mul_f32 + v_cvt_i32_f32 (trunc toward zero == jnp astype(int32);
    // |x*65536| < 2^22, no overflow).
    int xq = (int)(xf * 65536.0f);

    // seg = #{ bp_i <= xq }  via binary search (4 compares, 3 bpermutes).
    int seg = (xq >= bp7) ? 8 : 0;
    int m = __builtin_amdgcn_ds_bpermute((seg + 3) << 2, bpl);   // bp[3|11]
    seg += (xq >= m) ? 4 : 0;
    m = __builtin_amdgcn_ds_bpermute((seg + 1) << 2, bpl);       // bp[seg+1]
    seg += (xq >= m) ? 2 : 0;
    m = __builtin_amdgcn_ds_bpermute(seg << 2, bpl);             // bp[seg]
    seg += (xq >= m) ? 1 : 0;

    // One ds_load_b64; entry = { b - sm, e | (sm & 0x80000000) }.
    int2 cw = tab[seg];
    int e  = cw.y & 31;        // exp in [0,8)
    int sm = cw.y >> 31;       // all-ones iff slope sign is -1
    int t  = (xq >> e) ^ sm;   // branchless +/- (the "-sm" is folded into b)
    // i32->f32 rounds to nearest (== astype(float32)); *2^-16 exact.
    return (float)(cw.x + t) * INV_SCALE;
}

__global__ __launch_bounds__(256) void plac_kernel(
    const float* __restrict__ x,
    const int* __restrict__ g_bp,   // 15 sorted breakpoints (Q16.16)
    const int* __restrict__ g_ic,   // 16 intercepts (Q16.16)
    const int* __restrict__ g_sg,   // 16 signs in {-1,+1}
    const int* __restrict__ g_ex,   // 16 slope exponents in [0,8)
    float* __restrict__ out,
    int n) {
    // 128B LDS coefficient table.
    __shared__ int2 tab[NSEG];
    if (threadIdx.x < NSEG) {
        int b  = g_ic[threadIdx.x];
        int e  = g_ex[threadIdx.x];
        int s  = g_sg[threadIdx.x];
        int sm = (s < 0) ? -1 : 0;
        tab[threadIdx.x] = make_int2(b - sm, e | (sm & (int)0x80000000u));
    }
    __syncthreads();

    // Per-lane breakpoint register (bpermute source): lanes 0..14 hold the
    // breakpoints; higher lanes are never addressed (indices <= 14).
    const int lane = threadIdx.x & 31;
    const int bpl  = g_bp[lane < NBP ? lane : NBP - 1];
    // Uniform constant-index load -> SGPR operand for the first compare.
    const int bp7  = g_bp[7];

    const int n4     = n >> 2;
    const int tid    = blockIdx.x * blockDim.x + threadIdx.x;
    const int stride = gridDim.x * blockDim.x;
    const v4f* __restrict__ xv = (const v4f*)x;
    v4f* __restrict__ ov = (v4f*)out;

    // Wave-coherent grid-stride loop: the trip condition uses the wave's
    // first-lane index (runtime-uniform), so EXEC stays all-ones through the
    // ds_bpermutes. Load index is clamped; only the store is predicated.
    for (int v = tid, wb = tid - lane; wb < n4; v += stride, wb += stride) {
        int vc = v < n4 ? v : n4 - 1;
        v4f a = __builtin_nontemporal_load(xv + vc);   // global_load_b128 NT
        v4f r;
#pragma unroll
        for (int k = 0; k < 4; ++k) r[k] = plac_fast(a[k], bp7, bpl, tab);
        if (v < n4) __builtin_nontemporal_store(r, ov + vc);  // b128 NT
    }

    // Tail for n % 4 (dormant for n = 2^27): divergent, so no cross-lane
    // ops here -- plain scan with direct global coefficient reads.
    const int rem = n & 3;
    if (tid < rem) {
        const int base = n & ~3;
        int xq  = (int)(x[base + tid] * 65536.0f);
        int seg = 0;
#pragma unroll
        for (int i = 0; i < NBP; ++i) seg += (xq >= g_bp[i]) ? 1 : 0;
        int y = g_ic[seg] + g_sg[seg] * (xq >> g_ex[seg]);
        out[base + tid] = (float)y * INV_SCALE;
    }
}

extern "C" void kernel_launch(void* const* d_in, const int* in_sizes, int n_in,
                              void* d_out, int out_size, void* d_ws, size_t ws_size,
                              hipStream_t stream) {
    const float* x   = (const float*)d_in[0];  // (32, 4096, 1024) f32
    const int*  g_bp = (const int*)d_in[1];    // 15
    const int*  g_ic = (const int*)d_in[2];    // 16
    const int*  g_sg = (const int*)d_in[3];    // 16
    const int*  g_ex = (const int*)d_in[4];    // 16
    float* out = (float*)d_out;

    const int n  = in_sizes[0];
    const int n4 = n >> 2;

    // 8192 blocks x 256 threads = 64K wave32s; ~16 float4 iters/thread so
    // the table/bpermute-source setup amortizes.
    int blocks = (n4 + 255) / 256;
    if (blocks > 8192) blocks = 8192;
    if (blocks < 1) blocks = 1;

    plac_kernel<<<blocks, 256, 0, stream>>>(x, g_bp, g_ic, g_sg, g_ex, out, n);
}